// DifferentiableSolver_12575664243147
// MI455X (gfx1250) — compile-verified
//
#include <hip/hip_runtime.h>
#include <cstdint>

#define BIGF        1000000000.0f
#define GRID_N      128
#define BATCH       4
#define N_SRC       4
#define N_RCV       16
#define NUM_SWEEPS  512
#define MAX_TRACE   512
#define CELLS       (GRID_N * GRID_N)   // 16384 cells = 64KB f32

typedef unsigned int u32x4 __attribute__((ext_vector_type(4)));
typedef int          i32x4 __attribute__((ext_vector_type(4)));
typedef int          i32x8 __attribute__((ext_vector_type(8)));

// ---------------------------------------------------------------------------
// CDNA5 async global->LDS copy (ASYNCcnt path), inline asm (portable).
// LDS byte address = low 32 bits of the generic __shared__ pointer
// (flat aperture rule).
// ---------------------------------------------------------------------------
__device__ __forceinline__ void async_load_f32_to_lds(const float* gptr, const float* lds_ptr) {
    uint32_t lds_off = (uint32_t)(uintptr_t)lds_ptr;
    uint64_t gaddr   = (uint64_t)(uintptr_t)gptr;
    asm volatile("global_load_async_to_lds_b32 %0, %1, off"
                 :: "v"(lds_off), "v"(gaddr)
                 : "memory");
}

__device__ __forceinline__ void wait_async_zero() {
#if __has_builtin(__builtin_amdgcn_s_wait_asynccnt)
    __builtin_amdgcn_s_wait_asynccnt(0);
#else
    asm volatile("s_wait_asynccnt 0" ::: "memory");
#endif
}

__device__ __forceinline__ void wait_tensor_zero() {
#if __has_builtin(__builtin_amdgcn_s_wait_tensorcnt)
    __builtin_amdgcn_s_wait_tensorcnt(0);
#else
    asm volatile("s_wait_tensorcnt 0" ::: "memory");
#endif
}

// ---------------------------------------------------------------------------
// TDM: one descriptor-driven DMA of a 128x128 f32 tile (global -> LDS).
// D# built per CDNA5 ISA §8.3/8.4: group0 = {flags, lds_addr, global_addr,
// type=2}; group1 = {data_size=4B, tensor_dim0/1=128, tile_dim0/1=128,
// tensor_dim0_stride=128}. Issued wave-uniform; tracked by TENSORcnt.
// ---------------------------------------------------------------------------
#if __has_builtin(__builtin_amdgcn_tensor_load_to_lds)
#define HAVE_TDM 1
__device__ __forceinline__ void tdm_load_tile_128x128(const float* gptr, const float* lds_ptr) {
    uint64_t g   = (uint64_t)(uintptr_t)gptr;
    uint32_t glo = (uint32_t)__builtin_amdgcn_readfirstlane((int)(uint32_t)g);
    uint32_t ghi = (uint32_t)__builtin_amdgcn_readfirstlane((int)(uint32_t)(g >> 32));
    uint32_t lds = (uint32_t)__builtin_amdgcn_readfirstlane((int)(uint32_t)(uintptr_t)lds_ptr);

    u32x4 g0 = 0;
    g0[0] = 1u;                                     // count=1, user-mode load
    g0[1] = lds;                                    // lds_addr (bytes)
    g0[2] = glo;                                    // global_addr[31:0]
    g0[3] = (ghi & 0x01FFFFFFu) | (2u << 30);       // global_addr[56:32] | type=2

    i32x8 g1 = 0;
    g1[0] = (int)(2u << 16);                        // data_size=2 (4 bytes), wg_mask=0
    g1[1] = (int)((unsigned)GRID_N << 16);          // tensor_dim0[15:0] @ bits63:48
    g1[2] = (int)((unsigned)GRID_N << 16);          // tensor_dim1[15:0] @ bits95:80
    g1[3] = (int)((unsigned)GRID_N << 16);          // tile_dim0 @ bits127:112
    g1[4] = (int)GRID_N;                            // tile_dim1 @ bits143:128
    g1[5] = (int)GRID_N;                            // tensor_dim0_stride[31:0] @ bits191:160

    i32x4 z4 = 0;
#if __has_include(<hip/amd_detail/amd_gfx1250_TDM.h>)
    i32x8 z8 = 0;                                   // therock toolchain: 6-arg form
    __builtin_amdgcn_tensor_load_to_lds(g0, g1, z4, z4, z8, 0);
#else
    __builtin_amdgcn_tensor_load_to_lds(g0, g1, z4, z4, 0);   // ROCm 7.2: 5-arg form
#endif
}
#endif

// ---------------------------------------------------------------------------
// Kernel 1: eikonal Jacobi fixed-point, one workgroup per (b,s) grid.
// T grid lives entirely in LDS (64KB << 320KB/WGP). Thread tid owns the
// vertical strip: column j = tid&127, rows [strip*16, strip*16+16) where
// strip = tid>>7. Up/down neighbors come from registers (monotone-safe
// Gauss-Seidel ordering within the strip); only left/right + 2 strip-end
// values touch LDS: 50 conflict-free b32 LDS ops/thread/sweep vs 80 for the
// naive mapping. 1024 threads = 32 wave32 waves.
// ---------------------------------------------------------------------------
__global__ __launch_bounds__(1024)
void eikonal_kernel(const float* __restrict__ sos,
                    const int*   __restrict__ sources,
                    float*       __restrict__ Tout) {
    __shared__ float Tl[CELLS];

    const int tid   = threadIdx.x;
    const int b     = blockIdx.x >> 2;     // / N_SRC
    const int s     = blockIdx.x & 3;      // % N_SRC
    const int j     = tid & (GRID_N - 1);  // column
    const int strip = tid >> 7;            // 0..7
    const int base  = (strip * 16) * GRID_N + j;
    constexpr int K = 16;                  // rows per strip

    // --- stage sos[b] tile into LDS ---
    const float* sb = sos + (size_t)b * CELLS;
#ifdef HAVE_TDM
    if (tid < 32) {                        // wave-uniform: wave 0 issues the DMA
        tdm_load_tile_128x128(sb, &Tl[0]);
    }
    wait_tensor_zero();
#else
    #pragma unroll
    for (int k = 0; k < K; ++k) {
        const int c = tid + k * 1024;
        async_load_f32_to_lds(&sb[c], &Tl[c]);
    }
    wait_async_zero();
#endif
    __syncthreads();

    // dt = 1/max(speed,1e-12) (speed>0 else BIG), kept in registers
    float dt[K], Ts[K];
    #pragma unroll
    for (int k = 0; k < K; ++k) {
        const float sp = Tl[base + k * GRID_N];
        dt[k] = (sp > 0.0f) ? (1.0f / fmaxf(sp, 1e-12f)) : BIGF;
    }
    __syncthreads();   // everyone done reading sos before LDS is reused for T

    // --- initialize T: BIG everywhere, 0 at the source cell ---
    const int si = sources[2 * s + 0];
    const int sj = sources[2 * s + 1];
    const int srcIdx = si * GRID_N + sj;
    #pragma unroll
    for (int k = 0; k < K; ++k) {
        const int c = base + k * GRID_N;
        Ts[k] = (c == srcIdx) ? 0.0f : BIGF;
        Tl[c] = Ts[k];
    }
    __syncthreads();

    const bool has_lf = (j > 0);
    const bool has_rt = (j < GRID_N - 1);
    const bool has_up = (strip > 0);
    const bool has_dn = (strip < 7);

    // --- 512 sweeps, fully LDS-resident ---
    for (int sweep = 0; sweep < NUM_SWEEPS; ++sweep) {
        // strip-end neighbor rows (clamped unconditional loads + select)
        float up_b = Tl[has_up ? (base - GRID_N) : base];
        float dn_b = Tl[has_dn ? (base + K * GRID_N) : base];
        up_b = has_up ? up_b : BIGF;
        dn_b = has_dn ? dn_b : BIGF;

        #pragma unroll
        for (int k = 0; k < K; ++k) {
            const int c = base + k * GRID_N;
            const float up = (k == 0)     ? up_b : Ts[k - 1];   // register-carried
            const float dn = (k == K - 1) ? dn_b : Ts[k + 1];   // register-carried

            float lf = Tl[has_lf ? (c - 1) : c];
            float rt = Tl[has_rt ? (c + 1) : c];
            lf = has_lf ? lf : BIGF;
            rt = has_rt ? rt : BIGF;

            const float tx = fminf(up, dn);
            const float ty = fminf(lf, rt);
            const bool fx = tx < BIGF;
            const bool fy = ty < BIGF;
            const float txs = fx ? tx : 0.0f;
            const float tys = fy ? ty : 0.0f;
            const float d    = dt[k];
            const float diff = txs - tys;
            const float disc = 2.0f * d * d - diff * diff;
            const float sq   = (disc > 0.0f) ? __builtin_sqrtf(fmaxf(disc, 1e-30f)) : 0.0f;
            const float quad = (disc >= 0.0f) ? 0.5f * (txs + tys + sq) : BIGF;
            const float tent = (fx && fy) ? quad
                             : (fx ? (txs + d)
                             : (fy ? (tys + d) : BIGF));
            const float nv = fminf(Ts[k], tent);
            Ts[k] = nv;
            Tl[c] = nv;
        }
        __syncthreads();
    }

    // --- write solved grid to workspace (authoritative copy is in registers) ---
    float* to = Tout + (size_t)blockIdx.x * CELLS;
    #pragma unroll
    for (int k = 0; k < K; ++k) {
        to[base + k * GRID_N] = Ts[k];     // lane-consecutive => coalesced
    }
}

// ---------------------------------------------------------------------------
// Kernel 2: fill output with +inf (reference: tof = full(inf))
// ---------------------------------------------------------------------------
__global__ void fill_inf_kernel(float* __restrict__ out, int n) {
    const int idx = blockIdx.x * blockDim.x + threadIdx.x;
    if (idx < n) out[idx] = __builtin_inff();
}

// ---------------------------------------------------------------------------
// Kernel 3: backtrace. One block per (b,s); async-stage the solved 64KB grid
// into LDS (per-lane ASYNCcnt path), then 16 lanes (one per receiver) walk
// 512 dependent steps out of LDS and scatter results into out[b, s, r].
// ---------------------------------------------------------------------------
__global__ __launch_bounds__(256)
void backtrace_kernel(const float* __restrict__ T,
                      const int*   __restrict__ sources,
                      const int*   __restrict__ receivers,
                      float*       __restrict__ out) {
    __shared__ float Tl[CELLS];

    const int tid = threadIdx.x;
    const int b   = blockIdx.x >> 2;
    const int s   = blockIdx.x & 3;

    const float* Tg = T + (size_t)blockIdx.x * CELLS;
    #pragma unroll 4
    for (int k = 0; k < CELLS / 256; ++k) {
        const int c = tid + k * 256;
        async_load_f32_to_lds(&Tg[c], &Tl[c]);
    }
    wait_async_zero();
    __syncthreads();

    if (tid < N_RCV) {
        const int r  = tid;
        const int ei = sources[2 * s + 0];
        const int ej = sources[2 * s + 1];
        int i = receivers[2 * r + 0];
        int j = receivers[2 * r + 1];
        float t   = 0.0f;
        bool done = (i == ei) && (j == ej);

        for (int step = 0; step < MAX_TRACE; ++step) {
            // candidate order matches reference stacking: i-1, i+1, j-1, j+1
            const int ci[4] = { i - 1, i + 1, i,     i     };
            const int cj[4] = { j,     j,     j - 1, j + 1 };
            float best = 2.0f * BIGF;   // > BIG so first-min semantics of argmin hold
            int ni = i, nj = j;
            #pragma unroll
            for (int k = 0; k < 4; ++k) {
                const int ii = ci[k], jj = cj[k];
                const bool valid = (ii >= 0) && (ii < GRID_N) && (jj >= 0) && (jj < GRID_N);
                const int ic = ii < 0 ? 0 : (ii > GRID_N - 1 ? GRID_N - 1 : ii);
                const int jc = jj < 0 ? 0 : (jj > GRID_N - 1 ? GRID_N - 1 : jj);
                float tv = Tl[ic * GRID_N + jc];
                tv = valid ? tv : BIGF;
                if (tv < best) { best = tv; ni = ii; nj = jj; }   // strict < => first min
            }
            if (!done) { i = ni; j = nj; t += best; }
            done = done || ((i == ei) && (j == ej));
        }
        out[(size_t)b * CELLS + s * GRID_N + r] = t;
    }
}

// ---------------------------------------------------------------------------
// Launch: eikonal (16 WGs) -> fill +inf -> backtrace (16 WGs), all on `stream`.
// Workspace holds T[B,S,128,128] = 1 MB of f32.
// ---------------------------------------------------------------------------
extern "C" void kernel_launch(void* const* d_in, const int* in_sizes, int n_in,
                              void* d_out, int out_size, void* d_ws, size_t ws_size,
                              hipStream_t stream) {
    (void)in_sizes; (void)n_in; (void)out_size; (void)ws_size;
    const float* sos       = (const float*)d_in[0];
    const int*   sources   = (const int*)d_in[1];
    const int*   receivers = (const int*)d_in[2];
    float*       out       = (float*)d_out;
    float*       Tws       = (float*)d_ws;   // BATCH*N_SRC*CELLS floats = 1 MB

    eikonal_kernel<<<BATCH * N_SRC, 1024, 0, stream>>>(sos, sources, Tws);
    fill_inf_kernel<<<(BATCH * CELLS + 255) / 256, 256, 0, stream>>>(out, BATCH * CELLS);
    backtrace_kernel<<<BATCH * N_SRC, 256, 0, stream>>>(Tws, sources, receivers, out);
}